// LogLinearMarkovWithBaseline_46694884442577
// MI455X (gfx1250) — compile-verified
//
#include <hip/hip_runtime.h>

typedef __attribute__((ext_vector_type(2))) float v2f;
typedef __attribute__((ext_vector_type(8))) float v8f;

#define NSTATES 128
#define UDIM 4

// ---------------- workspace layout (bytes) ----------------
// Wfull  : float4[128*128]          @ 0        (262144)
// counts : int[128]                 @ 262144   (512)
// binOff : int[129]                 @ 262656   (pad to 640)
// tileOff: int[129]                 @ 263296   (pad to 640)
// cursor : int[128]                 @ 263936   (512)
// sorted : int[T]                   @ 264448
#define OFF_WFULL   0
#define OFF_COUNTS  262144
#define OFF_BINOFF  262656
#define OFF_TILEOFF 263296
#define OFF_CURSOR  263936
#define OFF_SORTED  264448

// ---------------------------------------------------------------------------
// K0: build dense Wfull[s][j][0..3] (self column zeroed) and zero counters.
// ---------------------------------------------------------------------------
__global__ void llm_build_kernel(const float* __restrict__ W,
                                 float4* __restrict__ Wfull,
                                 int* __restrict__ counts) {
    int idx = blockIdx.x * blockDim.x + threadIdx.x;
    if (idx < NSTATES) counts[idx] = 0;
    if (idx >= NSTATES * NSTATES) return;
    int s = idx >> 7;
    int j = idx & 127;
    float4 w = make_float4(0.f, 0.f, 0.f, 0.f);
    if (j != s) {
        int m = j - (j > s ? 1 : 0);              // inverse of the (N-1)->N map
        w = *(const float4*)(W + (size_t)(s * 127 + m) * UDIM);
    }
    Wfull[idx] = w;
}

// ---------------------------------------------------------------------------
// K1: histogram of x_curr with LDS aggregation.
// ---------------------------------------------------------------------------
__global__ void llm_hist_kernel(const int* __restrict__ x, int* __restrict__ counts, int T) {
    __shared__ int lc[NSTATES];
    int tid = threadIdx.x;
    if (tid < NSTATES) lc[tid] = 0;
    __syncthreads();
    int t = blockIdx.x * blockDim.x + tid;
    if (t < T) atomicAdd(&lc[x[t]], 1);
    __syncthreads();
    if (tid < NSTATES) {
        int v = lc[tid];
        if (v) atomicAdd(&counts[tid], v);
    }
}

// ---------------------------------------------------------------------------
// K2: serial 128-entry scan -> bin offsets, tile offsets, scatter cursors.
// ---------------------------------------------------------------------------
__global__ void llm_scan_kernel(const int* __restrict__ counts,
                                int* __restrict__ binOff,
                                int* __restrict__ tileOff,
                                int* __restrict__ cursor) {
    if (threadIdx.x == 0 && blockIdx.x == 0) {
        int acc = 0, tacc = 0;
        for (int s = 0; s < NSTATES; ++s) {
            binOff[s] = acc;
            cursor[s] = acc;
            tileOff[s] = tacc;
            acc  += counts[s];
            tacc += (counts[s] + 15) >> 4;
        }
        binOff[NSTATES] = acc;
        tileOff[NSTATES] = tacc;
    }
}

// ---------------------------------------------------------------------------
// K3: scatter timestep indices into state-sorted order.
// ---------------------------------------------------------------------------
__global__ void llm_scatter_kernel(const int* __restrict__ x,
                                   int* __restrict__ cursor,
                                   int* __restrict__ sorted, int T) {
    int t = blockIdx.x * blockDim.x + threadIdx.x;
    if (t >= T) return;
    int pos = atomicAdd(&cursor[x[t]], 1);
    sorted[pos] = t;
}

// ---------------------------------------------------------------------------
// K4: one wave per 16-timestep tile of one state bin.
//     logits(16x128) = WMMA_f32_16x16x4( U_tile , Wfull[s] ) + logP0[s]
//     then wave32 shfl-reduced row log-softmax, coalesced f32 stores.
// ---------------------------------------------------------------------------
__global__ __launch_bounds__(256) void llm_main_kernel(
        const float* __restrict__ u_curr,
        const float* __restrict__ logP0,
        const float4* __restrict__ Wfull,
        const int* __restrict__ binOff,
        const int* __restrict__ tileOff,
        const int* __restrict__ sorted,
        float* __restrict__ out, int T) {
    const int lane = threadIdx.x & 31;
    const int wv   = threadIdx.x >> 5;
    const int g    = blockIdx.x * 8 + wv;
    const int numTiles = tileOff[NSTATES];
    if (g >= numTiles) return;                     // uniform per wave -> EXEC stays full

    // binary search: s such that tileOff[s] <= g < tileOff[s+1]
    int lo = 0, hi = NSTATES;
    while (hi - lo > 1) {
        int mid = (lo + hi) >> 1;
        if (g >= tileOff[mid]) lo = mid; else hi = mid;
    }
    const int s = lo;
    const int lt       = g - tileOff[s];
    const int binE     = binOff[s + 1];
    const int rowStart = binOff[s] + lt * 16;

    const int  hl   = lane & 15;
    const bool hi16 = lane >= 16;

    // row owned by this lane for the A operand (lanes r and r+16 share row r)
    int rIdx = rowStart + hl;
    int tRow = sorted[min(rIdx, T - 1)];

    // A: 16x4 f32 -> lane<16: (K0,K1), lane>=16: (K2,K3)
    float2 av = *(const float2*)(u_curr + (size_t)tRow * UDIM + (hi16 ? 2 : 0));
    v2f a; a.x = av.x; a.y = av.y;

    const float4* wrow = Wfull + (size_t)s * NSTATES;
    const float*  lpr  = logP0 + (size_t)s * NSTATES;

    v8f acc[8];
#pragma unroll
    for (int jt = 0; jt < 8; ++jt) {
        // B: 4x16 f32 -> VGPR v holds row K = v + 2*hi16, col = lane&15
        const float* bp = (const float*)(wrow + jt * 16 + hl) + (hi16 ? 2 : 0);
        float2 bv = *(const float2*)bp;
        v2f b; b.x = bv.x; b.y = bv.y;
        // C pre-filled with broadcast logP0 (column-only term) -> WMMA fuses baseline add
        float lp = lpr[jt * 16 + hl];
        v8f ci;
#pragma unroll
        for (int k = 0; k < 8; ++k) ci[k] = lp;
        acc[jt] = __builtin_amdgcn_wmma_f32_16x16x4_f32(
            false, a, false, b, (short)0, ci, false, false);
    }

    // ---- row-wise log-softmax; row M = v + 8*hi16 lives in acc[*][v] ----
    float lse[8];
#pragma unroll
    for (int v = 0; v < 8; ++v) {
        float mv = acc[0][v];
#pragma unroll
        for (int jt = 1; jt < 8; ++jt) mv = fmaxf(mv, acc[jt][v]);
#pragma unroll
        for (int d = 1; d < 16; d <<= 1)           // reduce within 16-lane half
            mv = fmaxf(mv, __shfl_xor(mv, d, 32));
        float sv = 0.f;
#pragma unroll
        for (int jt = 0; jt < 8; ++jt) sv += __expf(acc[jt][v] - mv);
#pragma unroll
        for (int d = 1; d < 16; d <<= 1)
            sv += __shfl_xor(sv, d, 32);
        lse[v] = mv + __logf(sv);
    }

    // ---- store: row M's timestep fetched from lane M via shuffle ----
#pragma unroll
    for (int v = 0; v < 8; ++v) {
        int M  = v + (hi16 ? 8 : 0);
        int rI = rowStart + M;
        int tM = __shfl(tRow, M, 32);
        if (rI < binE) {
            float* op = out + (size_t)tM * NSTATES + hl;
#pragma unroll
            for (int jt = 0; jt < 8; ++jt)
                op[jt * 16] = acc[jt][v] - lse[v];  // 64B contiguous per half-wave
        }
    }
}

// ---------------------------------------------------------------------------
// Fallback (workspace too small): one wave per t, pure VALU.
// ---------------------------------------------------------------------------
__global__ __launch_bounds__(256) void llm_plain_kernel(
        const int* __restrict__ x_curr, const float* __restrict__ u_curr,
        const float* __restrict__ logP0, const float* __restrict__ W,
        float* __restrict__ out, int T) {
    int lane = threadIdx.x & 31;
    int t = blockIdx.x * 8 + (threadIdx.x >> 5);
    if (t >= T) return;
    int x = x_curr[t];
    float4 u = *(const float4*)(u_curr + (size_t)t * UDIM);
    float4 lp = *(const float4*)(logP0 + (size_t)x * NSTATES + lane * 4);
    float lg[4];
    const float* lpp = &lp.x;
#pragma unroll
    for (int i = 0; i < 4; ++i) {
        int j = lane * 4 + i;
        float st = 0.f;
        if (j != x) {
            int m = j - (j > x ? 1 : 0);
            float4 w = *(const float4*)(W + (size_t)(x * 127 + m) * UDIM);
            st = u.x * w.x + u.y * w.y + u.z * w.z + u.w * w.w;
        }
        lg[i] = lpp[i] + st;
    }
    float mv = fmaxf(fmaxf(lg[0], lg[1]), fmaxf(lg[2], lg[3]));
#pragma unroll
    for (int d = 1; d < 32; d <<= 1) mv = fmaxf(mv, __shfl_xor(mv, d, 32));
    float sv = __expf(lg[0] - mv) + __expf(lg[1] - mv) + __expf(lg[2] - mv) + __expf(lg[3] - mv);
#pragma unroll
    for (int d = 1; d < 32; d <<= 1) sv += __shfl_xor(sv, d, 32);
    float lse = mv + __logf(sv);
    float4 o = make_float4(lg[0] - lse, lg[1] - lse, lg[2] - lse, lg[3] - lse);
    *(float4*)(out + (size_t)t * NSTATES + lane * 4) = o;
}

extern "C" void kernel_launch(void* const* d_in, const int* in_sizes, int n_in,
                              void* d_out, int out_size, void* d_ws, size_t ws_size,
                              hipStream_t stream) {
    const int*   x_curr = (const int*)d_in[0];
    const float* u_curr = (const float*)d_in[1];
    const float* logP0  = (const float*)d_in[2];
    const float* W      = (const float*)d_in[3];
    float*       out    = (float*)d_out;
    const int T = in_sizes[0];

    const size_t wsNeeded = (size_t)OFF_SORTED + (size_t)T * sizeof(int);
    if (d_ws != nullptr && ws_size >= wsNeeded) {
        char* ws = (char*)d_ws;
        float4* Wfull  = (float4*)(ws + OFF_WFULL);
        int* counts    = (int*)(ws + OFF_COUNTS);
        int* binOff    = (int*)(ws + OFF_BINOFF);
        int* tileOff   = (int*)(ws + OFF_TILEOFF);
        int* cursor    = (int*)(ws + OFF_CURSOR);
        int* sorted    = (int*)(ws + OFF_SORTED);

        llm_build_kernel<<<(NSTATES * NSTATES + 255) / 256, 256, 0, stream>>>((const float*)W, Wfull, counts);
        llm_hist_kernel<<<(T + 255) / 256, 256, 0, stream>>>(x_curr, counts, T);
        llm_scan_kernel<<<1, 1, 0, stream>>>(counts, binOff, tileOff, cursor);
        llm_scatter_kernel<<<(T + 255) / 256, 256, 0, stream>>>(x_curr, cursor, sorted, T);

        int maxTiles = (T + 15) / 16 + NSTATES;    // upper bound; kernel checks tileOff[128]
        llm_main_kernel<<<(maxTiles + 7) / 8, 256, 0, stream>>>(
            u_curr, logP0, Wfull, binOff, tileOff, sorted, out, T);
    } else {
        llm_plain_kernel<<<(T + 7) / 8, 256, 0, stream>>>(x_curr, u_curr, logP0, W, out, T);
    }
}